// UniSAGELayer_19327352832459
// MI455X (gfx1250) — compile-verified
//
#include <hip/hip_runtime.h>

typedef __attribute__((ext_vector_type(2))) float v2f;
typedef __attribute__((ext_vector_type(8))) float v8f;

#define N_NODES 100000
#define N_EDGES 20000
#define NNZ     1600000
#define D       128

// -------------------------------------------------------------------------
// h = x_0 @ W^T + b  via V_WMMA_F32_16X16X4_F32 (exact fp32, bias in C init)
// One wave per 16x16 output tile; 8 waves/block cover the 8 N-tiles (128/16)
// of one 16-row M-tile. Grid = N_NODES/16 = 6250 blocks.
// -------------------------------------------------------------------------
__global__ __launch_bounds__(256) void unisage_gemm_wmma(
    const float* __restrict__ x0, const float* __restrict__ W,
    const float* __restrict__ b, float* __restrict__ h)
{
    const int wave    = threadIdx.x >> 5;   // 0..7 -> which 16-col tile
    const int lane    = threadIdx.x & 31;
    const int lrow    = lane & 15;          // M (for A) / N (for B,C,D)
    const int khalf   = (lane >> 4) << 1;   // 0 or 2: which K-pair this half-wave holds
    const int row_base = blockIdx.x << 4;   // node row of this tile
    const int n_base   = wave << 4;         // output-channel base

    // C layout: lane holds column N = n_base + lrow for all 8 VGPRs -> bias
    const float bias = b[n_base + lrow];
    v8f c;
#pragma unroll
    for (int i = 0; i < 8; ++i) c[i] = bias;

    // A: row (row_base + lrow) of x0;  B[k][n] = W[n][k] -> row n of W
    const float* arow = x0 + (size_t)(row_base + lrow) * D;
    const float* wrow = W  + (size_t)(n_base  + lrow) * D;

#pragma unroll
    for (int k = 0; k < D; k += 4) {
        v2f a  = *(const v2f*)(arow + k + khalf);   // K = k+khalf, k+khalf+1
        v2f bb = *(const v2f*)(wrow + k + khalf);
        // 8 args: (neg_a, A, neg_b, B, c_mod, C, reuse_a, reuse_b)
        c = __builtin_amdgcn_wmma_f32_16x16x4_f32(
                false, a, false, bb, (short)0, c, false, false);
    }

    // D layout: VGPR v -> M = v + 8*(lane>>4), N = lrow
    const int mhi = (lane >> 4) << 3;
#pragma unroll
    for (int v = 0; v < 8; ++v)
        h[(size_t)(row_base + v + mhi) * D + n_base + lrow] = c[v];
}

// -------------------------------------------------------------------------
// Hop 1: m_0_1[edge] += h[node]  (one wave32 per nnz; 32 lanes x float4 = 128)
// Also accumulates deg[node] (float) for the later mean.
// -------------------------------------------------------------------------
__global__ __launch_bounds__(256) void unisage_scatter_edges(
    const float* __restrict__ h, const int* __restrict__ node_idx,
    const int* __restrict__ edge_idx, float* __restrict__ m01,
    float* __restrict__ deg)
{
    const int i = blockIdx.x * 8 + (threadIdx.x >> 5);  // nnz index
    if (i >= NNZ) return;
    const int lane = threadIdx.x & 31;
    const int n = node_idx[i];
    const int e = edge_idx[i];

    const float4 val = ((const float4*)(h + (size_t)n * D))[lane];
    float* dst = m01 + (size_t)e * D + lane * 4;
    atomicAdd(dst + 0, val.x);
    atomicAdd(dst + 1, val.y);
    atomicAdd(dst + 2, val.z);
    atomicAdd(dst + 3, val.w);
    if (lane == 0) atomicAdd(deg + n, 1.0f);
}

// -------------------------------------------------------------------------
// Hop 2: msg[node] += m_0_1[edge]
// -------------------------------------------------------------------------
__global__ __launch_bounds__(256) void unisage_scatter_nodes(
    const float* __restrict__ m01, const int* __restrict__ node_idx,
    const int* __restrict__ edge_idx, float* __restrict__ msg)
{
    const int i = blockIdx.x * 8 + (threadIdx.x >> 5);
    if (i >= NNZ) return;
    const int lane = threadIdx.x & 31;
    const int n = node_idx[i];
    const int e = edge_idx[i];

    const float4 val = ((const float4*)(m01 + (size_t)e * D))[lane];
    float* dst = msg + (size_t)n * D + lane * 4;
    atomicAdd(dst + 0, val.x);
    atomicAdd(dst + 1, val.y);
    atomicAdd(dst + 2, val.z);
    atomicAdd(dst + 3, val.w);
}

// -------------------------------------------------------------------------
// out[n,c] = h[n,c] + msg[n,c] / max(deg[n], 1)
// (out already holds h; empty rows -> msg==0 -> +0, matching reference)
// -------------------------------------------------------------------------
__global__ __launch_bounds__(256) void unisage_finalize(
    float* __restrict__ out, const float* __restrict__ msg,
    const float* __restrict__ deg)
{
    const int idx = blockIdx.x * 256 + threadIdx.x;
    if (idx >= N_NODES * D) return;
    const int n = idx >> 7;  // /128
    const float d = fmaxf(deg[n], 1.0f);
    out[idx] += msg[idx] / d;
}

extern "C" void kernel_launch(void* const* d_in, const int* in_sizes, int n_in,
                              void* d_out, int out_size, void* d_ws, size_t ws_size,
                              hipStream_t stream)
{
    const float* x0       = (const float*)d_in[0];
    const float* W        = (const float*)d_in[1];
    const float* b        = (const float*)d_in[2];
    const int*   node_idx = (const int*)d_in[3];
    const int*   edge_idx = (const int*)d_in[4];
    float*       out      = (float*)d_out;

    // workspace layout: m01 [20000*128] | msg [100000*128] | deg [100000]
    float* m01 = (float*)d_ws;
    float* msg = m01 + (size_t)N_EDGES * D;
    float* deg = msg + (size_t)N_NODES * D;
    const size_t zero_bytes =
        ((size_t)N_EDGES * D + (size_t)N_NODES * D + (size_t)N_NODES) * sizeof(float);

    hipMemsetAsync(d_ws, 0, zero_bytes, stream);

    unisage_gemm_wmma<<<N_NODES / 16, 256, 0, stream>>>(x0, W, b, out);
    unisage_scatter_edges<<<NNZ / 8, 256, 0, stream>>>(out, node_idx, edge_idx, m01, deg);
    unisage_scatter_nodes<<<NNZ / 8, 256, 0, stream>>>(m01, node_idx, edge_idx, msg);
    unisage_finalize<<<(N_NODES * D) / 256, 256, 0, stream>>>(out, msg, deg);
}